// SageLayer_2954937500236
// MI455X (gfx1250) — compile-verified
//
#include <hip/hip_runtime.h>

#define NN      50000
#define KNEIGH  16
#define IN_DIM  128
#define OUT_DIM 128
#define CDIM    256        // 2*IN_DIM
#define LDS_ROW 260        // padded row (floats): 4-float pad -> conflict-free LDS reads
#define M_TILE  16

typedef __attribute__((ext_vector_type(2))) float v2f;
typedef __attribute__((ext_vector_type(4))) float v4f;
typedef __attribute__((ext_vector_type(8))) float v8f;

// 16B async global->LDS copy (ASYNCcnt-tracked). lds_off = byte offset into
// the workgroup's LDS segment (dynamic LDS starts at 0: no static __shared__).
__device__ __forceinline__ void async_copy_b128(unsigned lds_off, const void* gptr) {
    unsigned long long ga = (unsigned long long)(uintptr_t)gptr;
    asm volatile("global_load_async_to_lds_b128 %0, %1, off"
                 :: "v"(lds_off), "v"(ga) : "memory");
}

__device__ __forceinline__ void wait_async0() {
    asm volatile("s_wait_asynccnt 0" ::: "memory");
}

__global__ __launch_bounds__(256) void sage_fused_kernel(
    const int*   __restrict__ nodes,
    const float* __restrict__ embs,
    const int*   __restrict__ neigh,
    const float* __restrict__ weight,
    float*       __restrict__ out)
{
    extern __shared__ char smem[];
    float* combined = (float*)smem;                              // [16][260]
    int*   s_nodes  = (int*)(smem + M_TILE * LDS_ROW * 4);       // [16]
    int*   s_neigh  = s_nodes + M_TILE;                          // [16*16]

    const int tid  = threadIdx.x;
    const int base = blockIdx.x * M_TILE;

    // ---- stage 0: indices to LDS -------------------------------------
    if (tid < M_TILE) s_nodes[tid] = nodes[base + tid];
    s_neigh[tid] = neigh[base * KNEIGH + tid];                   // 256 = 16*16
    __syncthreads();

    // stage-2 wave geometry computed early so we can prefetch weights now
    const int wave  = tid >> 5;
    const int lane  = tid & 31;
    const int nloc  = lane & 15;
    const int khalf = lane >> 4;
    const int n0    = wave * 16;
    const float* wrow = weight + (size_t)(n0 + nloc) * CDIM + 2 * khalf;
    __builtin_prefetch(wrow, 0, 0);          // global_prefetch_b8: stream W into cache
    __builtin_prefetch(wrow + 256, 0, 0);

    // ---- stage 1: build combined[16][256] in LDS ---------------------
    const int quad  = tid & 31;              // float4 slot: 32 quads cover 128 dims
    const int mrow0 = tid >> 5;              // 0..7 ; handle rows m and m+8
#pragma unroll
    for (int rep = 0; rep < 2; ++rep) {
        const int m = mrow0 + rep * 8;
        // self features: async DMA copy global -> combined[m][0..127]
        const unsigned lds_off = (unsigned)((m * LDS_ROW + quad * 4) * 4);
        const float* self_src = embs + (size_t)s_nodes[m] * IN_DIM + quad * 4;
        async_copy_b128(lds_off, self_src);

        // neighbor mean: accumulate 16 gathered float4's in registers
        v4f acc = {0.f, 0.f, 0.f, 0.f};
#pragma unroll
        for (int j = 0; j < KNEIGH; ++j) {
            const v4f* p = (const v4f*)(embs + (size_t)s_neigh[m * KNEIGH + j] * IN_DIM + quad * 4);
            acc += *p;
        }
        *(v4f*)(combined + m * LDS_ROW + IN_DIM + quad * 4) = acc * 0.0625f;
    }
    wait_async0();            // our own async writes to LDS done
    __syncthreads();          // all waves' combined tile visible

    // ---- stage 2: [16 x 256] @ W^T tile via V_WMMA_F32_16X16X4_F32 ----
    // A 16x4 f32: lanes 0-15 M=0..15 K={0,1}, lanes 16-31 M=0..15 K={2,3}
    // B 4x16 f32: symmetric layout, N=lane%16, K base = 2*khalf
    const float* arow = combined + nloc * LDS_ROW + 2 * khalf;
    v8f acc = {};
#pragma unroll 8
    for (int kk = 0; kk < CDIM / 4; ++kk) {
        v2f a = *(const v2f*)(arow + kk * 4);     // ds_load_b64, bank-conflict-free
        v2f b = *(const v2f*)(wrow + kk * 4);     // global_load_b64, L2/L0 hot
        acc = __builtin_amdgcn_wmma_f32_16x16x4_f32(
            /*neg_a=*/false, a, /*neg_b=*/false, b,
            /*c_mod=*/(short)0, acc, /*reuse_a=*/false, /*reuse_b=*/false);
    }

    // ---- store with fused ReLU ---------------------------------------
    // C/D 16x16 f32 layout: VGPR v -> row (khalf*8 + v), column = n0 + nloc
#pragma unroll
    for (int v = 0; v < 8; ++v) {
        const int m = khalf * 8 + v;
        float x = acc[v];
        out[(size_t)(base + m) * OUT_DIM + n0 + nloc] = x > 0.f ? x : 0.f;
    }
}

extern "C" void kernel_launch(void* const* d_in, const int* in_sizes, int n_in,
                              void* d_out, int out_size, void* d_ws, size_t ws_size,
                              hipStream_t stream) {
    const int*   nodes  = (const int*)d_in[0];
    const float* embs   = (const float*)d_in[1];
    const int*   neigh  = (const int*)d_in[2];
    const float* weight = (const float*)d_in[3];
    float*       out    = (float*)d_out;

    const int    blocks = NN / M_TILE;                                  // 3125
    const size_t lds    = (size_t)M_TILE * LDS_ROW * 4 + (M_TILE + M_TILE * KNEIGH) * 4;
    sage_fused_kernel<<<dim3(blocks), dim3(256), lds, stream>>>(nodes, embs, neigh, weight, out);
}